// WeatherPrediction_42477226557667
// MI455X (gfx1250) — compile-verified
//
#include <hip/hip_runtime.h>
#include <stdint.h>

typedef __attribute__((ext_vector_type(2))) float v2f;
typedef __attribute__((ext_vector_type(4))) float v4f;
typedef __attribute__((ext_vector_type(8))) float v8f;

#define NGRID 65160
#define NMESH 10242
#define NTOT  75402
#define FIN   156
#define FEXT  3
#define HDIM  128
#define FOUTD 78
#define SOUT  80
#define BATCH 2
#define EENC  130320
#define EPROC 81920
#define EDEC  195480

// ---------------------------------------------------------------------------
// Generic GEMM: C[b,n,0:Fout] = (reluA? relu(A[b,n,:]) : A[b,n,:]) @ W + bias
// A rows have stride K=128 (BATCH==2 assumed: b = row>=R).
// WT is W transposed: WT[c*128 + k]. Block = 64 rows staged in LDS; 8 waves
// each own one 16-col tile and carry 4 accumulators (4 row tiles), so each
// B fragment (one global_load_b64) feeds 4 v_wmma_f32_16x16x4_f32.
// CS (output row stride) is a template constant so epilogue stores use
// immediate offsets from one base address.
// ---------------------------------------------------------------------------
template<int CS>
__global__ __launch_bounds__(256)
void gemm128_wmma(const float* __restrict__ A, long aBatchStride,
                  const float* __restrict__ WT, const float* __restrict__ bias,
                  float* __restrict__ C,
                  int R, int totalRows, int Fout, int reluA)
{
    const int K = 128, KS = 132, ROWS = 64;
    __shared__ __align__(16) float As[ROWS * KS];
    int tile0 = blockIdx.x * ROWS;

    // stage 64 x 128 A-tile (b128 loads, b128 LDS stores)
    for (int q = threadIdx.x; q < ROWS * (K / 4); q += 256) {
        int mm = q >> 5;
        int kq = (q & 31) << 2;
        int row = tile0 + mm;
        v4f v = {0.f, 0.f, 0.f, 0.f};
        if (row < totalRows) {
            int b = (row >= R) ? 1 : 0;
            int n = row - b * R;
            v = *(const v4f*)(A + (long)b * aBatchStride + (long)n * K + kq);
            if (reluA) {
                v.x = fmaxf(v.x, 0.f); v.y = fmaxf(v.y, 0.f);
                v.z = fmaxf(v.z, 0.f); v.w = fmaxf(v.w, 0.f);
            }
        }
        *(v4f*)&As[mm * KS + kq] = v;
    }
    __syncthreads();

    int wv = threadIdx.x >> 5, lane = threadIdx.x & 31;
    int colTiles = (Fout + 15) >> 4;
    if (wv >= colTiles) return;            // uniform per-wave; no later syncs

    int m = lane & 15, half = lane >> 4;
    int col  = (wv << 4) + m;
    int colc = (col < Fout) ? col : (Fout - 1);   // clamp: no divergent loads
    const float* wt = WT + (long)colc * K + 2 * half;
    const float* ap[4];
#pragma unroll
    for (int r = 0; r < 4; ++r) ap[r] = &As[(r * 16 + m) * KS + 2 * half];

    v8f acc[4];
#pragma unroll
    for (int r = 0; r < 4; ++r) acc[r] = (v8f){0.f,0.f,0.f,0.f,0.f,0.f,0.f,0.f};

    for (int k0 = 0; k0 < K; k0 += 4) {
        v2f b = *(const v2f*)(wt + k0);    // one b64 load, shared by 4 WMMAs
#pragma unroll
        for (int r = 0; r < 4; ++r) {
            v2f a;
            a.x = ap[r][k0];
            a.y = ap[r][k0 + 1];
            acc[r] = __builtin_amdgcn_wmma_f32_16x16x4_f32(false, a, false, b,
                                                           (short)0, acc[r],
                                                           false, false);
        }
    }

    float bv = bias[colc];
    float* cp = C + (long)(tile0 + 8 * half) * CS + col;   // per-lane base
    bool fullRows = (tile0 + ROWS) <= totalRows;           // uniform
    bool fullCols = ((wv << 4) + 16) <= Fout;              // uniform per wave
    if (fullRows && fullCols) {
#pragma unroll
        for (int r = 0; r < 4; ++r)
#pragma unroll
            for (int v = 0; v < 8; ++v)
                cp[(r * 16 + v) * CS] = acc[r][v] + bv;    // immediate offsets
    } else {
        bool cok = col < Fout;
#pragma unroll
        for (int r = 0; r < 4; ++r)
#pragma unroll
            for (int v = 0; v < 8; ++v) {
                int row = tile0 + r * 16 + v + 8 * half;
                if (row < totalRows && cok)
                    cp[(r * 16 + v) * CS] = acc[r][v] + bv;
            }
    }
}

// ---------------------------------------------------------------------------
// Layer-1 GEMM with fused input gather: Xall = [X | init_feats], K=159 pad 160.
// WT1 is transposed w1 with zero-padded k=159 row: WT1[c*160 + k].
// ---------------------------------------------------------------------------
__global__ __launch_bounds__(256)
void gemm_enc_in_wmma(const float* __restrict__ X, const float* __restrict__ gF,
                      const float* __restrict__ mF, const float* __restrict__ WT1,
                      const float* __restrict__ bias, float* __restrict__ C)
{
    const int KP = 160, KS = 164, KA = 159, ROWS = 64;
    __shared__ __align__(16) float As[ROWS * KS];
    int tile0 = blockIdx.x * ROWS;
    const int totalRows = BATCH * NTOT;

    for (int idx = threadIdx.x; idx < ROWS * KP; idx += 256) {
        int mm = idx / KP, kk = idx - mm * KP;
        int row = tile0 + mm;
        float v = 0.f;
        if (row < totalRows && kk < KA) {
            int b = (row >= NTOT) ? 1 : 0;
            int n = row - b * NTOT;
            if (kk < FIN) {
                v = (n < NGRID) ? X[((long)b * NGRID + n) * FIN + kk] : 0.f;
            } else {
                v = (n < NGRID) ? gF[(long)n * FEXT + (kk - FIN)]
                                : mF[(long)(n - NGRID) * FEXT + (kk - FIN)];
            }
        }
        As[mm * KS + kk] = v;
    }
    __syncthreads();

    int wv = threadIdx.x >> 5, lane = threadIdx.x & 31;
    int m = lane & 15, half = lane >> 4;
    int col = (wv << 4) + m;
    const float* wt = WT1 + (long)col * KP + 2 * half;
    const float* ap[4];
#pragma unroll
    for (int r = 0; r < 4; ++r) ap[r] = &As[(r * 16 + m) * KS + 2 * half];

    v8f acc[4];
#pragma unroll
    for (int r = 0; r < 4; ++r) acc[r] = (v8f){0.f,0.f,0.f,0.f,0.f,0.f,0.f,0.f};

    for (int k0 = 0; k0 < KP; k0 += 4) {
        v2f b = *(const v2f*)(wt + k0);    // pad row k=159 is zero-filled
#pragma unroll
        for (int r = 0; r < 4; ++r) {
            v2f a;
            a.x = ap[r][k0];
            a.y = ap[r][k0 + 1];
            acc[r] = __builtin_amdgcn_wmma_f32_16x16x4_f32(false, a, false, b,
                                                           (short)0, acc[r],
                                                           false, false);
        }
    }

    float bv = bias[col];
    float* cp = C + (long)(tile0 + 8 * half) * HDIM + col;
    if ((tile0 + ROWS) <= totalRows) {
#pragma unroll
        for (int r = 0; r < 4; ++r)
#pragma unroll
            for (int v = 0; v < 8; ++v)
                cp[(r * 16 + v) * HDIM] = acc[r][v] + bv;
    } else {
#pragma unroll
        for (int r = 0; r < 4; ++r)
#pragma unroll
            for (int v = 0; v < 8; ++v) {
                int row = tile0 + r * 16 + v + 8 * half;
                if (row < totalRows)
                    cp[(r * 16 + v) * HDIM] = acc[r][v] + bv;
            }
    }
}

// ---------------------------------------------------------------------------
// small helpers
// ---------------------------------------------------------------------------
__global__ void fill_kernel(float* a, float val, int n) {
    int i = blockIdx.x * blockDim.x + threadIdx.x;
    if (i < n) a[i] = val;
}
__global__ void deg_count_kernel(float* deg, const int* __restrict__ dst, int E) {
    int i = blockIdx.x * blockDim.x + threadIdx.x;
    if (i < E) atomicAdd(&deg[dst[i]], 1.0f);
}
__global__ void rsqrt_kernel(float* a, int n) {
    int i = blockIdx.x * blockDim.x + threadIdx.x;
    if (i < n) a[i] = rsqrtf(a[i]);
}
// WT[c*KT + k] = W[k*Fout + c]
__global__ void transpose_w_kernel(const float* __restrict__ W, float* __restrict__ WT,
                                   int K, int Fout, int KT) {
    int i = blockIdx.x * blockDim.x + threadIdx.x;
    if (i >= K * Fout) return;
    int k = i / Fout, c = i - k * Fout;
    WT[(long)c * KT + k] = W[i];
}

// out[row,f] = XW[row,f] * dinv[n]^2 + bias[f]   (self-loop + bias init)
// grid.x = 2*R rows, block = 128 threads
__global__ void gcn_self_bias_kernel(const float* __restrict__ XW,
                                     const float* __restrict__ dinv,
                                     const float* __restrict__ bias,
                                     float* __restrict__ out,
                                     int R, int F, int SX, int SO)
{
    int row = blockIdx.x;
    int f = threadIdx.x;
    if (f >= F) return;
    int b = (row >= R) ? 1 : 0;
    int n = row - b * R;
    float dv = dinv[n];
    out[(long)row * SO + f] = XW[(long)row * SX + f] * dv * dv + bias[f];
}

// out[b,dst,:] += XW[b,src,:] * dinv[src]*dinv[dst]; wave per edge, grid.y = batch
__global__ void gcn_scatter_kernel(const float* __restrict__ XW,
                                   const int* __restrict__ src,
                                   const int* __restrict__ dst,
                                   const float* __restrict__ dinv,
                                   float* __restrict__ out,
                                   int E, int R, int F, int SX, int SO)
{
    int e = (blockIdx.x * blockDim.x + threadIdx.x) >> 5;
    int lane = threadIdx.x & 31;
    int b = blockIdx.y;
    if (e >= E) return;
    int s = src[e], d = dst[e];
    float nm = dinv[s] * dinv[d];
    const float* xs = XW + ((long)b * R + s) * SX;
    float*       od = out + ((long)b * R + d) * SO;
    for (int f = lane; f < F; f += 32)
        atomicAdd(&od[f], xs[f] * nm);
}

// mesh rows of pf <- processed mesh features; grid (NMESH, 2), block 128
__global__ void copy_mesh_kernel(const float* __restrict__ M, float* __restrict__ pf) {
    int n = blockIdx.x, b = blockIdx.y, f = threadIdx.x;
    pf[((long)b * NTOT + NGRID + n) * HDIM + f] = M[((long)b * NMESH + n) * HDIM + f];
}
// d_out[b,n,0:78] = agg[b,n,0:78] (stride 80); grid (NGRID, 2), block 128
__global__ void final_copy_kernel(const float* __restrict__ agg, float* __restrict__ out) {
    int n = blockIdx.x, b = blockIdx.y, f = threadIdx.x;
    if (f >= FOUTD) return;
    out[((long)b * NGRID + n) * FOUTD + f] = agg[((long)b * NTOT + n) * SOUT + f];
}

// ---------------------------------------------------------------------------
extern "C" void kernel_launch(void* const* d_in, const int* in_sizes, int n_in,
                              void* d_out, int out_size, void* d_ws, size_t ws_size,
                              hipStream_t stream)
{
    (void)in_sizes; (void)n_in; (void)out_size; (void)ws_size;
    const float* X        = (const float*)d_in[0];
    const float* gridF    = (const float*)d_in[1];
    const float* meshF    = (const float*)d_in[2];
    const float* e_mlp_w1 = (const float*)d_in[3];
    const float* e_mlp_b1 = (const float*)d_in[4];
    const float* e_mlp_w2 = (const float*)d_in[5];
    const float* e_mlp_b2 = (const float*)d_in[6];
    const float* e_gcn_w1 = (const float*)d_in[7];
    const float* e_gcn_b1 = (const float*)d_in[8];
    const float* e_gcn_w2 = (const float*)d_in[9];
    const float* e_gcn_b2 = (const float*)d_in[10];
    const float* p_gcn_w1 = (const float*)d_in[11];
    const float* p_gcn_b1 = (const float*)d_in[12];
    const float* p_gcn_w2 = (const float*)d_in[13];
    const float* p_gcn_b2 = (const float*)d_in[14];
    const float* d_gcn_w1 = (const float*)d_in[15];
    const float* d_gcn_b1 = (const float*)d_in[16];
    const float* d_gcn_w2 = (const float*)d_in[17];
    const float* d_gcn_b2 = (const float*)d_in[18];
    const int*   enc_src  = (const int*)d_in[19];
    const int*   enc_dst  = (const int*)d_in[20];
    const int*   proc_src = (const int*)d_in[21];
    const int*   proc_dst = (const int*)d_in[22];
    const int*   dec_src  = (const int*)d_in[23];
    const int*   dec_dst  = (const int*)d_in[24];
    float* out = (float*)d_out;

    char* ws = (char*)d_ws;
    const size_t BIG = (size_t)BATCH * NTOT * HDIM * sizeof(float);   // ~77.2 MB
    float* buf0 = (float*)(ws);
    float* buf1 = (float*)(ws + BIG);
    float* buf2 = (float*)(ws + 2 * BIG);
    float* dinvEnc  = (float*)(ws + 3 * BIG);
    float* dinvDec  = dinvEnc + NTOT;
    float* dinvProc = dinvDec + NTOT;
    float* WT_mlp1  = dinvProc + NMESH;        // 128 cols x 160 stride (zero-pad k=159)
    float* WT_mlp2  = WT_mlp1 + 128 * 160;
    float* WT_eg1   = WT_mlp2 + 128 * 128;
    float* WT_eg2   = WT_eg1  + 128 * 128;
    float* WT_pg1   = WT_eg2  + 128 * 128;
    float* WT_pg2   = WT_pg1  + 128 * 128;
    float* WT_dg1   = WT_pg2  + 128 * 128;
    float* WT_dg2   = WT_dg1  + 128 * 128;     // 78 cols x 128 stride
    const size_t MESHF = (size_t)BATCH * NMESH * HDIM;
    float* M1 = buf0 + MESHF;
    float* M3 = buf2 + MESHF;

    const int totalN = BATCH * NTOT;   // 150804
    const int totalM = BATCH * NMESH;  // 20484
    const int gN = (totalN + 63) / 64;
    const int gM = (totalM + 63) / 64;

    // ---- weight transposes ------------------------------------------------
    fill_kernel<<<(128 * 160 + 255) / 256, 256, 0, stream>>>(WT_mlp1, 0.f, 128 * 160);
    transpose_w_kernel<<<(159 * 128 + 255) / 256, 256, 0, stream>>>(e_mlp_w1, WT_mlp1, 159, 128, 160);
    transpose_w_kernel<<<(128 * 128 + 255) / 256, 256, 0, stream>>>(e_mlp_w2, WT_mlp2, 128, 128, 128);
    transpose_w_kernel<<<(128 * 128 + 255) / 256, 256, 0, stream>>>(e_gcn_w1, WT_eg1, 128, 128, 128);
    transpose_w_kernel<<<(128 * 128 + 255) / 256, 256, 0, stream>>>(e_gcn_w2, WT_eg2, 128, 128, 128);
    transpose_w_kernel<<<(128 * 128 + 255) / 256, 256, 0, stream>>>(p_gcn_w1, WT_pg1, 128, 128, 128);
    transpose_w_kernel<<<(128 * 128 + 255) / 256, 256, 0, stream>>>(p_gcn_w2, WT_pg2, 128, 128, 128);
    transpose_w_kernel<<<(128 * 128 + 255) / 256, 256, 0, stream>>>(d_gcn_w1, WT_dg1, 128, 128, 128);
    transpose_w_kernel<<<(128 * 78 + 255) / 256, 256, 0, stream>>>(d_gcn_w2, WT_dg2, 128, 78, 128);

    // ---- degree / dinv for the three graphs -------------------------------
    fill_kernel<<<(NTOT + 255) / 256, 256, 0, stream>>>(dinvEnc, 1.f, NTOT);
    fill_kernel<<<(NTOT + 255) / 256, 256, 0, stream>>>(dinvDec, 1.f, NTOT);
    fill_kernel<<<(NMESH + 255) / 256, 256, 0, stream>>>(dinvProc, 1.f, NMESH);
    deg_count_kernel<<<(EENC + 255) / 256, 256, 0, stream>>>(dinvEnc, enc_dst, EENC);
    deg_count_kernel<<<(EDEC + 255) / 256, 256, 0, stream>>>(dinvDec, dec_dst, EDEC);
    deg_count_kernel<<<(EPROC + 255) / 256, 256, 0, stream>>>(dinvProc, proc_dst, EPROC);
    rsqrt_kernel<<<(NTOT + 255) / 256, 256, 0, stream>>>(dinvEnc, NTOT);
    rsqrt_kernel<<<(NTOT + 255) / 256, 256, 0, stream>>>(dinvDec, NTOT);
    rsqrt_kernel<<<(NMESH + 255) / 256, 256, 0, stream>>>(dinvProc, NMESH);

    const dim3 egE((EENC * 32 + 255) / 256, BATCH);
    const dim3 egP((EPROC * 32 + 255) / 256, BATCH);
    const dim3 egD((EDEC * 32 + 255) / 256, BATCH);

    // ---- 1) hidden = Xall @ w1 + b1 -> buf0 -------------------------------
    gemm_enc_in_wmma<<<gN, 256, 0, stream>>>(X, gridF, meshF, WT_mlp1, e_mlp_b1, buf0);
    // ---- 2) h = relu(hidden) @ w2 + b2 -> buf1 ----------------------------
    gemm128_wmma<HDIM><<<gN, 256, 0, stream>>>(buf0, (long)NTOT * HDIM, WT_mlp2, e_mlp_b2,
                                               buf1, NTOT, totalN, HDIM, 1);
    // ---- 3) enc conv1: XW -> buf0; agg -> buf2 ----------------------------
    gemm128_wmma<HDIM><<<gN, 256, 0, stream>>>(buf1, (long)NTOT * HDIM, WT_eg1, e_gcn_b1,
                                               buf0, NTOT, totalN, HDIM, 0);
    gcn_self_bias_kernel<<<totalN, 128, 0, stream>>>(buf0, dinvEnc, e_gcn_b1, buf2,
                                                     NTOT, HDIM, HDIM, HDIM);
    gcn_scatter_kernel<<<egE, 256, 0, stream>>>(buf0, enc_src, enc_dst, dinvEnc, buf2,
                                                EENC, NTOT, HDIM, HDIM, HDIM);
    // ---- 4) enc conv2: input relu(buf2); agg -> buf1 (= h) ----------------
    gemm128_wmma<HDIM><<<gN, 256, 0, stream>>>(buf2, (long)NTOT * HDIM, WT_eg2, e_gcn_b2,
                                               buf0, NTOT, totalN, HDIM, 1);
    gcn_self_bias_kernel<<<totalN, 128, 0, stream>>>(buf0, dinvEnc, e_gcn_b2, buf1,
                                                     NTOT, HDIM, HDIM, HDIM);
    gcn_scatter_kernel<<<egE, 256, 0, stream>>>(buf0, enc_src, enc_dst, dinvEnc, buf1,
                                                EENC, NTOT, HDIM, HDIM, HDIM);
    // ---- 5) proc conv1 on mesh rows of buf1 -> buf0; agg -> M1 ------------
    gemm128_wmma<HDIM><<<gM, 256, 0, stream>>>(buf1 + (size_t)NGRID * HDIM, (long)NTOT * HDIM,
                                               WT_pg1, p_gcn_b1,
                                               buf0, NMESH, totalM, HDIM, 0);
    gcn_self_bias_kernel<<<totalM, 128, 0, stream>>>(buf0, dinvProc, p_gcn_b1, M1,
                                                     NMESH, HDIM, HDIM, HDIM);
    gcn_scatter_kernel<<<egP, 256, 0, stream>>>(buf0, proc_src, proc_dst, dinvProc, M1,
                                                EPROC, NMESH, HDIM, HDIM, HDIM);
    // ---- 6) proc conv2: input relu(M1) -> buf2; agg -> M3 -----------------
    gemm128_wmma<HDIM><<<gM, 256, 0, stream>>>(M1, (long)NMESH * HDIM, WT_pg2, p_gcn_b2,
                                               buf2, NMESH, totalM, HDIM, 1);
    gcn_self_bias_kernel<<<totalM, 128, 0, stream>>>(buf2, dinvProc, p_gcn_b2, M3,
                                                     NMESH, HDIM, HDIM, HDIM);
    gcn_scatter_kernel<<<egP, 256, 0, stream>>>(buf2, proc_src, proc_dst, dinvProc, M3,
                                                EPROC, NMESH, HDIM, HDIM, HDIM);
    // ---- 7) pf = [grid_h | m]: copy M3 into mesh rows of buf1 -------------
    {
        dim3 g(NMESH, BATCH);
        copy_mesh_kernel<<<g, 128, 0, stream>>>(M3, buf1);
    }
    // ---- 8) dec conv1: XW -> buf0, agg -> buf2 ----------------------------
    gemm128_wmma<HDIM><<<gN, 256, 0, stream>>>(buf1, (long)NTOT * HDIM, WT_dg1, d_gcn_b1,
                                               buf0, NTOT, totalN, HDIM, 0);
    gcn_self_bias_kernel<<<totalN, 128, 0, stream>>>(buf0, dinvDec, d_gcn_b1, buf2,
                                                     NTOT, HDIM, HDIM, HDIM);
    gcn_scatter_kernel<<<egD, 256, 0, stream>>>(buf0, dec_src, dec_dst, dinvDec, buf2,
                                                EDEC, NTOT, HDIM, HDIM, HDIM);
    // ---- 9) dec conv2: input relu(buf2), Fout=78 (stride 80) --------------
    gemm128_wmma<SOUT><<<gN, 256, 0, stream>>>(buf2, (long)NTOT * HDIM, WT_dg2, d_gcn_b2,
                                               buf0, NTOT, totalN, FOUTD, 1);
    gcn_self_bias_kernel<<<totalN, 128, 0, stream>>>(buf0, dinvDec, d_gcn_b2, buf1,
                                                     NTOT, FOUTD, SOUT, SOUT);
    gcn_scatter_kernel<<<egD, 256, 0, stream>>>(buf0, dec_src, dec_dst, dinvDec, buf1,
                                                EDEC, NTOT, FOUTD, SOUT, SOUT);
    // ---- 10) strip grid rows -> d_out -------------------------------------
    {
        dim3 g(NGRID, BATCH);
        final_copy_kernel<<<g, 128, 0, stream>>>(buf1, out);
    }
}